// MultiScaleGNN_5059471475395
// MI455X (gfx1250) — compile-verified
//
#include <hip/hip_runtime.h>
#include <hip/hip_bf16.h>

typedef __attribute__((ext_vector_type(2))) float v2f;
typedef __attribute__((ext_vector_type(8))) float v8f;

constexpr int kN   = 20000;   // nodes
constexpr int kE   = 320000;  // edges
constexpr int kG   = 512;     // graphs
constexpr int kDIN = 2279;
constexpr int kD   = 512;
constexpr int kL   = 6;
constexpr int kS   = 3;
constexpr int kH   = 8;
constexpr int kT   = 13;

constexpr int BM = 128, BN = 128, BK = 16;
constexpr int LDK = BK + 2;   // even stride (18) -> 8B-aligned b64 LDS reads, conflict-free

// ---------------------------------------------------------------------------
// f32 WMMA GEMM:  C[M,N] = A[M,K] @ B[K,N] (+ bias[N]) (+ relu)
// 256 threads = 8 wave32; block tile 128x128; each wave owns a 16x128 strip
// (8 x 16x16 accumulators). A and B staged in LDS with K contiguous per row
// so every WMMA fragment is a single aligned ds_load_b64.
// ---------------------------------------------------------------------------
template<bool BIAS, bool RELU>
__global__ __launch_bounds__(256)
void wmma_gemm_f32(const float* __restrict__ A, const float* __restrict__ B,
                   const float* __restrict__ bias, float* __restrict__ C,
                   int M, int K, int N) {
    __shared__ float As[BM][LDK];   // row-major rows of A  (K contiguous)
    __shared__ float Bs[BN][LDK];   // column-major cols of B (K contiguous)

    const int tid   = threadIdx.x;
    const int wave  = tid >> 5;        // 0..7
    const int lane  = tid & 31;
    const int half  = lane >> 4;       // 0/1
    const int lrow  = lane & 15;
    const int rowBase = blockIdx.y * BM;
    const int colBase = blockIdx.x * BN;
    const int wr = wave * 16;          // wave's row offset inside block tile

    v8f zero = {};
    v8f acc[8];
#pragma unroll
    for (int t = 0; t < 8; ++t) acc[t] = zero;

    for (int k0 = 0; k0 < K; k0 += BK) {
        // --- A tile: 128x16 = 2048 elts, 8 per thread (coalesced along K) ---
#pragma unroll
        for (int i = 0; i < (BM * BK) / 256; ++i) {
            int idx = tid + i * 256;
            int r = idx >> 4, c = idx & 15;
            int gr = rowBase + r, gc = k0 + c;
            As[r][c] = (gr < M && gc < K) ? A[(size_t)gr * K + gc] : 0.0f;
        }
        // --- B tile: 16x128 = 2048 elts, 8 per thread; stored K-contiguous ---
#pragma unroll
        for (int i = 0; i < (BK * BN) / 256; ++i) {
            int idx = tid + i * 256;
            int r = idx >> 7, c = idx & 127;            // r = K row, c = N col
            int gr = k0 + r, gc = colBase + c;
            Bs[c][r] = (gr < K && gc < N) ? B[(size_t)gr * N + gc] : 0.0f;
        }
        __syncthreads();

#pragma unroll
        for (int kk = 0; kk < BK; kk += 4) {
            // A 16x4 frag: a[i] = A[row][kk + i + 2*half]  -> one b64 read
            const v2f a = *(const v2f*)&As[wr + lrow][kk + 2 * half];
#pragma unroll
            for (int t = 0; t < 8; ++t) {
                // B 4x16 frag: b[i] = B[kk + i + 2*half][col] -> one b64 read
                const v2f b = *(const v2f*)&Bs[t * 16 + lrow][kk + 2 * half];
                acc[t] = __builtin_amdgcn_wmma_f32_16x16x4_f32(
                    false, a, false, b, (short)0, acc[t], false, false);
            }
        }
        __syncthreads();
    }

    // --- store: C/D layout: VGPR i -> row = i + 8*half, col = lane%16 ---
#pragma unroll
    for (int t = 0; t < 8; ++t) {
        int col = colBase + t * 16 + lrow;
        if (col < N) {
            float bv = BIAS ? bias[col] : 0.0f;
#pragma unroll
            for (int i = 0; i < 8; ++i) {
                int row = rowBase + wr + i + 8 * half;
                if (row < M) {
                    float v = acc[t][i] + bv;
                    if (RELU) v = fmaxf(v, 0.0f);
                    C[(size_t)row * N + col] = v;
                }
            }
        }
    }
}

// ---------------------------------------------------------------------------
// small helper kernels
// ---------------------------------------------------------------------------
__global__ void zero_kernel(float* __restrict__ p, int n) {
    int i = blockIdx.x * blockDim.x + threadIdx.x;
    if (i < n) p[i] = 0.0f;
}

__global__ void degree_kernel(const int* __restrict__ dst, float* __restrict__ deg) {
    int e = blockIdx.x * blockDim.x + threadIdx.x;
    if (e < kE) atomicAdd(&deg[dst[e]], 1.0f);
}

__global__ void rsqrt_kernel(float* __restrict__ dis) {
    int n = blockIdx.x * blockDim.x + threadIdx.x;
    if (n < kN) dis[n] = rsqrtf(dis[n] + 1.0f);   // deg = indeg + 1 (self loop)
}

__global__ void count_kernel(const int* __restrict__ batch, float* __restrict__ cnt) {
    int n = blockIdx.x * blockDim.x + threadIdx.x;
    if (n < kN) atomicAdd(&cnt[batch[n]], 1.0f);
}

// agg[n,d] = t[n,d] * dis[n]^2   (self-loop term, also initializes agg)
__global__ void self_scale_kernel(const float* __restrict__ t, const float* __restrict__ dis,
                                  float* __restrict__ agg) {
    int idx = blockIdx.x * blockDim.x + threadIdx.x;   // N*D = 10.24M
    if (idx < kN * kD) {
        int n = idx >> 9;   // D = 512
        float w = dis[n];
        agg[idx] = t[idx] * w * w;
    }
}

// agg[dst] += t[src] * dis[src]*dis[dst]   -- one block per edge (src/dst uniform
// per block -> scalar loads), 256 threads cover 512 feats in 2 steps.
__global__ void edge_scatter_kernel(const float* __restrict__ t, const int* __restrict__ src,
                                    const int* __restrict__ dst, const float* __restrict__ dis,
                                    float* __restrict__ agg) {
    int e = blockIdx.x;
    int s = src[e], d = dst[e];
    float w = dis[s] * dis[d];
    const float* ts = t + (size_t)s * kD;
    float*       ad = agg + (size_t)d * kD;
    for (int c = threadIdx.x; c < kD; c += blockDim.x)
        atomicAdd(&ad[c], ts[c] * w);
}

__global__ void bias_relu_kernel(const float* __restrict__ agg, const float* __restrict__ b,
                                 float* __restrict__ hs) {
    int idx = blockIdx.x * blockDim.x + threadIdx.x;
    if (idx < kN * kD) {
        int d = idx & (kD - 1);
        hs[idx] = fmaxf(agg[idx] + b[d], 0.0f);
    }
}

// comb[batch[n], s, d] += hs[n, d]
__global__ void pool_kernel(const float* __restrict__ hs, const int* __restrict__ batch,
                            float* __restrict__ comb, int s) {
    int idx = blockIdx.x * blockDim.x + threadIdx.x;
    if (idx < kN * kD) {
        int n = idx >> 9;
        int d = idx & (kD - 1);
        atomicAdd(&comb[(size_t)batch[n] * (kS * kD) + s * kD + d], hs[idx]);
    }
}

__global__ void div_cnt_kernel(float* __restrict__ comb, const float* __restrict__ cnt) {
    int idx = blockIdx.x * blockDim.x + threadIdx.x;
    if (idx < kG * kS * kD) {
        int g = idx / (kS * kD);
        comb[idx] /= fmaxf(cnt[g], 1.0f);
    }
}

// per-(graph, head, query-row) 3x3 softmax attention, Dh = 64
__global__ void attention_kernel(const float* __restrict__ q, const float* __restrict__ k,
                                 const float* __restrict__ v, float* __restrict__ o) {
    int tid = blockIdx.x * blockDim.x + threadIdx.x;
    if (tid >= kG * kS * kH) return;
    const int Dh = kD / kH;                 // 64
    int g  = tid / (kS * kH);
    int r  = tid % (kS * kH);
    int qi = r / kH;
    int hh = r % kH;

    const float* qp = q + ((size_t)(g * kS + qi)) * kD + hh * Dh;
    float sc[kS];
#pragma unroll
    for (int ki = 0; ki < kS; ++ki) {
        const float* kp = k + ((size_t)(g * kS + ki)) * kD + hh * Dh;
        float dot = 0.0f;
        for (int d = 0; d < Dh; ++d) dot += qp[d] * kp[d];
        sc[ki] = dot * 0.125f;              // 1/sqrt(64)
    }
    float m  = fmaxf(sc[0], fmaxf(sc[1], sc[2]));
    float e0 = expf(sc[0] - m), e1 = expf(sc[1] - m), e2 = expf(sc[2] - m);
    float inv = 1.0f / (e0 + e1 + e2);
    e0 *= inv; e1 *= inv; e2 *= inv;

    const float* v0 = v + ((size_t)(g * kS + 0)) * kD + hh * Dh;
    const float* v1 = v + ((size_t)(g * kS + 1)) * kD + hh * Dh;
    const float* v2 = v + ((size_t)(g * kS + 2)) * kD + hh * Dh;
    float* op = o + ((size_t)(g * kS + qi)) * kD + hh * Dh;
    for (int d = 0; d < Dh; ++d)
        op[d] = e0 * v0[d] + e1 * v1[d] + e2 * v2[d];
}

// ---------------------------------------------------------------------------
// host-side launch
// ---------------------------------------------------------------------------
static inline int ceildiv(int a, int b) { return (a + b - 1) / b; }

extern "C" void kernel_launch(void* const* d_in, const int* in_sizes, int n_in,
                              void* d_out, int out_size, void* d_ws, size_t ws_size,
                              hipStream_t stream) {
    (void)in_sizes; (void)n_in; (void)out_size; (void)ws_size;

    const float* x     = (const float*)d_in[0];
    const int*   ei    = (const int*)  d_in[1];
    const int*   src   = ei;
    const int*   dst   = ei + kE;
    const int*   batch = (const int*)  d_in[2];
    const float* w_in  = (const float*)d_in[3];
    const float* b_in  = (const float*)d_in[4];
    const float* gcn_w = (const float*)d_in[5];
    const float* gcn_b = (const float*)d_in[6];
    const float* wq = (const float*)d_in[7];  const float* bq = (const float*)d_in[8];
    const float* wk = (const float*)d_in[9];  const float* bk = (const float*)d_in[10];
    const float* wv = (const float*)d_in[11]; const float* bv = (const float*)d_in[12];
    const float* wo = (const float*)d_in[13]; const float* bo = (const float*)d_in[14];
    const float* w1 = (const float*)d_in[15]; const float* b1 = (const float*)d_in[16];
    const float* w2 = (const float*)d_in[17]; const float* b2 = (const float*)d_in[18];
    float* out = (float*)d_out;

    // workspace carve-out (256B aligned)
    char* wsp = (char*)d_ws;
    auto alloc = [&](size_t nfloats) {
        float* p = (float*)wsp;
        wsp += ((nfloats * sizeof(float) + 255) / 256) * 256;
        return p;
    };
    float* dis  = alloc(kN);
    float* cnt  = alloc(kG);
    float* h    = alloc((size_t)kN * kD);
    float* hs   = alloc((size_t)kN * kD);
    float* t    = alloc((size_t)kN * kD);
    float* agg  = alloc((size_t)kN * kD);
    float* comb = alloc((size_t)kG * kS * kD);
    // reuse after GCN phase: q->t, k->agg, v->hs, o->h, flat/oo->t, hidden->agg
    float* qb = t;  float* kb = agg;  float* vb = hs;  float* ob = h;
    float* oo = t;  float* hidden = agg;

    const int ND = kN * kD;               // 10,240,000
    const int TB = 256;

    auto gemm = [&](const float* A, const float* B, const float* bias, float* C,
                    int M, int K, int N, bool useBias, bool useRelu) {
        dim3 grid(ceildiv(N, BN), ceildiv(M, BM));
        if (useBias && useRelu)
            wmma_gemm_f32<true,  true ><<<grid, TB, 0, stream>>>(A, B, bias, C, M, K, N);
        else if (useBias)
            wmma_gemm_f32<true,  false><<<grid, TB, 0, stream>>>(A, B, bias, C, M, K, N);
        else
            wmma_gemm_f32<false, false><<<grid, TB, 0, stream>>>(A, B, bias, C, M, K, N);
    };

    // 1) normalization factors and graph sizes
    zero_kernel<<<ceildiv(kN, TB), TB, 0, stream>>>(dis, kN);
    zero_kernel<<<ceildiv(kG, TB), TB, 0, stream>>>(cnt, kG);
    zero_kernel<<<ceildiv(kG * kS * kD, TB), TB, 0, stream>>>(comb, kG * kS * kD);
    degree_kernel<<<ceildiv(kE, TB), TB, 0, stream>>>(dst, dis);
    rsqrt_kernel<<<ceildiv(kN, TB), TB, 0, stream>>>(dis);
    count_kernel<<<ceildiv(kN, TB), TB, 0, stream>>>(batch, cnt);

    // 2) input projection: h = x @ w_in + b_in   [20000 x 2279 x 512]
    gemm(x, w_in, b_in, h, kN, kDIN, kD, true, false);

    // 3) GCN stacks, 3 scales x 6 layers
    for (int s = 0; s < kS; ++s) {
        const float* in = h;
        for (int l = 0; l < kL; ++l) {
            const float* W = gcn_w + (size_t)(s * kL + l) * kD * kD;
            const float* b = gcn_b + (size_t)(s * kL + l) * kD;
            gemm(in, W, nullptr, t, kN, kD, kD, false, false);           // t = hs @ W
            self_scale_kernel<<<ceildiv(ND, TB), TB, 0, stream>>>(t, dis, agg);
            edge_scatter_kernel<<<kE, TB, 0, stream>>>(t, src, dst, dis, agg);
            bias_relu_kernel<<<ceildiv(ND, TB), TB, 0, stream>>>(agg, b, hs);
            in = hs;
        }
        pool_kernel<<<ceildiv(ND, TB), TB, 0, stream>>>(hs, batch, comb, s);
    }
    div_cnt_kernel<<<ceildiv(kG * kS * kD, TB), TB, 0, stream>>>(comb, cnt);

    // 4) multi-head attention over the 3 scales
    gemm(comb, wq, bq, qb, kG * kS, kD, kD, true, false);
    gemm(comb, wk, bk, kb, kG * kS, kD, kD, true, false);
    gemm(comb, wv, bv, vb, kG * kS, kD, kD, true, false);
    attention_kernel<<<ceildiv(kG * kS * kH, TB), TB, 0, stream>>>(qb, kb, vb, ob);
    gemm(ob, wo, bo, oo, kG * kS, kD, kD, true, false);                  // oo = o @ wo + bo

    // 5) head MLP: relu(flat @ w1 + b1) @ w2 + b2
    gemm(oo, w1, b1, hidden, kG, kS * kD, kD, true, true);               // [512 x 1536 x 512]
    gemm(hidden, w2, b2, out, kG, kD, kT, true, false);                  // [512 x 512 x 13]
}